// VectorQuantizer_19963007992473
// MI455X (gfx1250) — compile-verified
//
#include <hip/hip_runtime.h>
#include <math.h>

typedef float v2f __attribute__((ext_vector_type(2)));
typedef float v8f __attribute__((ext_vector_type(8)));

#define KC    1024          // codebook entries
#define DDIM  64            // channels / embedding dim
#define NB    16            // batch
#define HWSZ  4096          // 64*64 spatial
#define NTOT  (NB*HWSZ)     // 65536 rows
#define LSTR  68            // padded LDS row stride (floats): 16B aligned, conflict-free b64 reads
#define WAVES_PER_WG 8
#define TPB   (WAVES_PER_WG*32)

// LDS byte offset (wave-relative) of a generic pointer into shared memory
static __device__ __forceinline__ unsigned ldsOff(void* p) {
    return (unsigned)(unsigned long long)(__attribute__((address_space(3))) char*)p;
}

// CDNA5 async DMA: 16B global -> LDS, tracked by ASYNCcnt (no VGPR data path)
static __device__ __forceinline__ void asyncG2L_b128(unsigned lds_byte_off, const float* g) {
    asm volatile("global_load_async_to_lds_b128 %0, %1, off"
                 :: "v"(lds_byte_off), "v"((unsigned long long)(size_t)g)
                 : "memory");
}

// ---------------- kernel 0: e2 = ||emb_k||^2, zero accumulators ----------------
__global__ void vq_init(const float* __restrict__ emb, float* __restrict__ e2,
                        float* __restrict__ avgp, float* __restrict__ acc) {
    int k = blockIdx.x * blockDim.x + threadIdx.x;
    if (k < KC) {
        const float4* er = (const float4*)(emb + (size_t)k * DDIM);
        float s = 0.f;
#pragma unroll
        for (int i = 0; i < DDIM / 4; ++i) {
            float4 v = er[i];
            s += v.x * v.x + v.y * v.y + v.z * v.z + v.w * v.w;
        }
        e2[k] = s;
        avgp[k] = 0.f;
    }
    if (blockIdx.x == 0 && threadIdx.x < 8) acc[threadIdx.x] = 0.f;
}

// ------------- kernel 1: NCHW -> row-major normalized x[N,64], x2, 1/norm -------------
__global__ void vq_normalize(const float* __restrict__ in, float* __restrict__ x,
                             float* __restrict__ x2, float* __restrict__ rno) {
    int n = blockIdx.x * blockDim.x + threadIdx.x;     // n = b*4096 + hw
    int b = n >> 12, hw = n & (HWSZ - 1);
    const float* base = in + (size_t)b * DDIM * HWSZ + hw;
    float v[DDIM];
    float ss = 0.f;
#pragma unroll
    for (int c = 0; c < DDIM; ++c) { v[c] = base[(size_t)c * HWSZ]; ss += v[c] * v[c]; }
    float norm = sqrtf(ss);
    float rn = 1.f / fmaxf(norm, 1e-12f);
    float4* xr = (float4*)(x + (size_t)n * DDIM);
#pragma unroll
    for (int c4 = 0; c4 < DDIM / 4; ++c4) {
        float4 o;
        o.x = v[4 * c4 + 0] * rn; o.y = v[4 * c4 + 1] * rn;
        o.z = v[4 * c4 + 2] * rn; o.w = v[4 * c4 + 3] * rn;
        xr[c4] = o;
    }
    x2[n] = ss * rn * rn;   // == ||normalized x||^2 (faithful to reference)
    rno[n] = rn;
}

// ---------------- kernel 2: fused distances (WMMA) + softmax stats + argmin ----------------
// Each wave owns 16 rows; sweeps all 64 codeword tiles of 16 with V_WMMA_F32_16X16X4_F32.
__global__ __launch_bounds__(TPB) void vq_main(
        const float* __restrict__ xg, const float* __restrict__ x2g,
        const float* __restrict__ e2g, const float* __restrict__ embg,
        float* __restrict__ avgp_g, float* __restrict__ acc, int* __restrict__ idx_g) {
    extern __shared__ float smem[];
    float* lemb  = smem;                 // KC * LSTR floats (padded codebook)
    float* lavgp = smem + KC * LSTR;     // KC floats

    // stage codebook into LDS via CDNA5 async DMA (16B granules, padded layout)
    for (int i = threadIdx.x; i < KC * (DDIM / 4); i += blockDim.x) {   // uniform trip count
        int k = i >> 4, c4 = i & 15;
        asyncG2L_b128(ldsOff(&lemb[k * LSTR + 4 * c4]), embg + (size_t)i * 4);
    }
    for (int i = threadIdx.x; i < KC; i += blockDim.x) lavgp[i] = 0.f;
    asm volatile("s_wait_asynccnt 0x0" ::: "memory");
    __syncthreads();

    const int lane = threadIdx.x & 31;
    const int wave = threadIdx.x >> 5;
    const int half = lane >> 4;          // 0: K even pair / M 0..7 ; 1: K odd pair / M 8..15
    const int l15  = lane & 15;
    const int r0   = (blockIdx.x * WAVES_PER_WG + wave) * 16;

    // A tile: 16 rows x 64 ch, ISA layout: lane {M=l15}, VGPR pair holds ch 4*kc+2*half{,+1}
    v2f a[16];
    {
        const float* xrow = xg + (size_t)(r0 + l15) * DDIM + 2 * half;
#pragma unroll
        for (int kc = 0; kc < 16; ++kc) a[kc] = *(const v2f*)(xrow + 4 * kc);
    }
    float x2r[8];
#pragma unroll
    for (int j = 0; j < 8; ++j) x2r[j] = x2g[r0 + 8 * half + j];

    float m[8], S[8], T[8], bd[8];
    int bi[8];
#pragma unroll
    for (int j = 0; j < 8; ++j) { m[j] = -INFINITY; S[j] = 0.f; T[j] = 0.f; bd[j] = INFINITY; bi[j] = 0x7fffffff; }

    // ---- pass 1: streaming logsumexp (max, S, T=sum e^d * d) + argmin ----
    for (int t = 0; t < KC / 16; ++t) {
        int n0 = t * 16;
        float e2c = e2g[n0 + l15];
        const float* bbase = lemb + (n0 + l15) * LSTR + 2 * half;  // same striping as A
        v8f c = {};
#pragma unroll
        for (int kc = 0; kc < 16; ++kc) {
            v2f b = *(const v2f*)(bbase + 4 * kc);
            c = __builtin_amdgcn_wmma_f32_16x16x4_f32(false, a[kc], false, b,
                                                      (short)0, c, false, false);
        }
        int col = n0 + l15;
#pragma unroll
        for (int j = 0; j < 8; ++j) {
            float d = x2r[j] + e2c - 2.f * c[j];
            // single-exp streaming update: one of the two scales is exp(0)=1
            float diff = d - m[j];
            float ex = __expf(-fabsf(diff));
            bool newmax = diff > 0.f;
            float eo = newmax ? ex : 1.f;    // scale for old accumulators
            float ed = newmax ? 1.f : ex;    // weight for new element
            S[j] = S[j] * eo + ed;
            T[j] = T[j] * eo + ed * d;
            m[j] = fmaxf(m[j], d);
            bool better = (d < bd[j]) || (d == bd[j] && col < bi[j]);
            bd[j] = better ? d : bd[j];
            bi[j] = better ? col : bi[j];
        }
    }

    // butterfly reduce across the 16 lanes sharing each row (xor 1,2,4,8 stays in-half)
#pragma unroll
    for (int off = 1; off < 16; off <<= 1) {
#pragma unroll
        for (int j = 0; j < 8; ++j) {
            float mo = __shfl_xor(m[j], off, 32);
            float So = __shfl_xor(S[j], off, 32);
            float To = __shfl_xor(T[j], off, 32);
            float diff = mo - m[j];
            float ex = __expf(-fabsf(diff));
            float s0 = diff > 0.f ? ex : 1.f;   // scale self
            float s1 = diff > 0.f ? 1.f : ex;   // scale other
            S[j] = S[j] * s0 + So * s1;
            T[j] = T[j] * s0 + To * s1;
            m[j] = fmaxf(m[j], mo);
            float bdo = __shfl_xor(bd[j], off, 32);
            int   bio = __shfl_xor(bi[j], off, 32);
            bool better = (bdo < bd[j]) || (bdo == bd[j] && bio < bi[j]);
            bd[j] = better ? bdo : bd[j];
            bi[j] = better ? bio : bi[j];
        }
    }

    // one writer per row: idx + entropy H = logS + m - E[d]
    if (l15 < 8) {
        int j = l15;
        int r = r0 + 8 * half + j;
        idx_g[r] = bi[j];
        float H = __logf(S[j]) + m[j] - T[j] / S[j];
        atomicAdd(&acc[0], H);
    }

    // ---- pass 2: recompute distances, accumulate p into LDS avg_p ----
    float invS[8];
#pragma unroll
    for (int j = 0; j < 8; ++j) invS[j] = 1.f / S[j];
    for (int t = 0; t < KC / 16; ++t) {
        int n0 = t * 16;
        float e2c = e2g[n0 + l15];
        const float* bbase = lemb + (n0 + l15) * LSTR + 2 * half;
        v8f c = {};
#pragma unroll
        for (int kc = 0; kc < 16; ++kc) {
            v2f b = *(const v2f*)(bbase + 4 * kc);
            c = __builtin_amdgcn_wmma_f32_16x16x4_f32(false, a[kc], false, b,
                                                      (short)0, c, false, false);
        }
        float psum = 0.f;
#pragma unroll
        for (int j = 0; j < 8; ++j) {
            float d = x2r[j] + e2c - 2.f * c[j];
            psum += __expf(d - m[j]) * invS[j];
        }
        atomicAdd(&lavgp[n0 + l15], psum);   // ds_add_f32
    }
    __syncthreads();
    for (int i = threadIdx.x; i < KC; i += blockDim.x)
        atomicAdd(&avgp_g[i], lavgp[i]);
}

// ---------------- kernel 3: gather codebook, write NCHW output, accumulate MSE ----------------
__global__ void vq_epilogue(const float* __restrict__ in, const float* __restrict__ emb,
                            const float* __restrict__ rno, const int* __restrict__ idx_g,
                            float* __restrict__ out, float* __restrict__ acc) {
    int n = blockIdx.x * blockDim.x + threadIdx.x;
    int b = n >> 12, hw = n & (HWSZ - 1);
    int best = idx_g[n];
    float rn = rno[n];
    const float* er = emb + (size_t)best * DDIM;
    const float* ib = in + (size_t)b * DDIM * HWSZ + hw;
    float* ob = out + (size_t)b * DDIM * HWSZ + hw;
    float mse = 0.f;
#pragma unroll
    for (int c = 0; c < DDIM; ++c) {
        float e = er[c];
        float xv = ib[(size_t)c * HWSZ] * rn;
        float df = xv - e;
        mse += df * df;
        ob[(size_t)c * HWSZ] = e;            // straight-through forward value == quantized
    }
#pragma unroll
    for (int off = 16; off > 0; off >>= 1) mse += __shfl_xor(mse, off, 32);
    if ((threadIdx.x & 31) == 0) atomicAdd(&acc[1], mse);
}

// ---------------- kernel 4: finalize loss ----------------
__global__ void vq_finalize(const float* __restrict__ avgp_g, const float* __restrict__ acc,
                            float* __restrict__ loss_out) {
    __shared__ float red[256];
    float inter = 0.f;
    for (int k = threadIdx.x; k < KC; k += 256) {
        float ap = avgp_g[k] * (1.0f / (float)NTOT);
        inter += ap * __logf(ap + 1e-8f);    // inter_loss = +sum avg_p*log(avg_p+eps)
    }
    red[threadIdx.x] = inter;
    __syncthreads();
    for (int s = 128; s > 0; s >>= 1) {
        if (threadIdx.x < s) red[threadIdx.x] += red[threadIdx.x + s];
        __syncthreads();
    }
    if (threadIdx.x == 0) {
        float intra = acc[0] * (1.f / (float)NTOT);
        float mse = acc[1] * (1.f / ((float)NTOT * (float)DDIM));
        // q_latent + 0.25*e_latent (equal forward) + 0.1*(intra+inter)
        loss_out[0] = 1.25f * mse + 0.1f * (intra + red[0]);
    }
}

extern "C" void kernel_launch(void* const* d_in, const int* in_sizes, int n_in,
                              void* d_out, int out_size, void* d_ws, size_t ws_size,
                              hipStream_t stream) {
    const float* in  = (const float*)d_in[0];   // [16,64,64,64] NCHW
    const float* emb = (const float*)d_in[1];   // [1024,64]
    float* out = (float*)d_out;                 // [0]=loss, [1..]=quantized NCHW

    float* ws   = (float*)d_ws;
    float* x    = ws;                           // NTOT*64
    float* x2   = x + (size_t)NTOT * DDIM;      // NTOT
    float* rno  = x2 + NTOT;                    // NTOT
    float* e2   = rno + NTOT;                   // KC
    float* avgp = e2 + KC;                      // KC
    float* acc  = avgp + KC;                    // 8 scalars: [0]=intra_sum [1]=mse_sum
    int*   idx  = (int*)(acc + 8);              // NTOT ints

    vq_init<<<KC / 256, 256, 0, stream>>>(emb, e2, avgp, acc);
    vq_normalize<<<NTOT / 256, 256, 0, stream>>>(in, x, x2, rno);

    size_t smem = (size_t)(KC * LSTR + KC) * sizeof(float);   // ~276 KB of the 320 KB WGP LDS
    (void)hipFuncSetAttribute((const void*)vq_main,
                              hipFuncAttributeMaxDynamicSharedMemorySize, (int)smem);
    vq_main<<<NTOT / (16 * WAVES_PER_WG), TPB, smem, stream>>>(x, x2, e2, emb, avgp, acc, idx);

    vq_epilogue<<<NTOT / 256, 256, 0, stream>>>(in, emb, rno, idx, out + 1, acc);
    vq_finalize<<<1, 256, 0, stream>>>(avgp, acc, out);
}